// AdaptiveExpertScoreFusion_80161269613165
// MI455X (gfx1250) — compile-verified
//
#include <hip/hip_runtime.h>
#include <math.h>

typedef __attribute__((ext_vector_type(2))) float v2f;
typedef __attribute__((ext_vector_type(8))) float v8f;

#define N_B 512
#define L_B 1024
#define S_B 3
#define D_B 128
#define K_B 4
#define H_B 32
#define G_B 64

__device__ __forceinline__ float gelu_exact(float x) {
    return 0.5f * x * (1.0f + erff(x * 0.70710678118654752f));
}

// ---------------------------------------------------------------------------
// Kernel 1: per-n graph context mean + situation MLP -> sit_logits (N x K)
// Streams cust_repr (268MB) exactly once, fully coalesced (thread d reads
// consecutive d within a 512B line per iteration).
// ---------------------------------------------------------------------------
__global__ void __launch_bounds__(128) sit_gate_kernel(
    const float* __restrict__ veh, const float* __restrict__ cust,
    const float* __restrict__ sw1, const float* __restrict__ sb1,
    const float* __restrict__ sw2, const float* __restrict__ sb2,
    float* __restrict__ sit_out)
{
    __shared__ float ctx[D_B];
    __shared__ float vsh[D_B];
    __shared__ float hg[G_B];
    const int n = blockIdx.x;
    const int d = threadIdx.x;

    const float* cp = cust + (size_t)n * L_B * D_B + d;
    float acc = 0.f;
#pragma unroll 8
    for (int l = 0; l < L_B; ++l) acc += cp[(size_t)l * D_B];
    ctx[d] = acc * (1.0f / (float)L_B);
    vsh[d] = veh[n * D_B + d];
    __syncthreads();

    if (d < G_B) {
        float a = sb1[d];
        for (int i = 0; i < D_B; ++i) {
            a += vsh[i] * sw1[i * G_B + d];
            a += ctx[i] * sw1[(D_B + i) * G_B + d];
        }
        hg[d] = gelu_exact(a);
    }
    __syncthreads();

    if (d < K_B) {
        float a = sb2[d];
        for (int g = 0; g < G_B; ++g) a += hg[g] * sw2[g * K_B + d];
        sit_out[n * K_B + d] = a;
    }
}

// ---------------------------------------------------------------------------
// Kernel 2: fused candidate gating + expert MLP + softmax fusion.
// Each wave (32 lanes) owns 32 rows of the flattened (N*L) x D edge_emb and
// computes edge_emb @ edge_w with V_WMMA_F32_16X16X4_F32 (two 16x16 f32
// accumulators, 32 K-steps over D=128). edge_w lives in LDS transposed &
// zero-padded to 16 columns so no EXEC divergence is needed around WMMA.
// ---------------------------------------------------------------------------
__global__ void __launch_bounds__(256) fuse_kernel(
    const float* __restrict__ scores, const float* __restrict__ edge,
    const float* __restrict__ ew,
    const float* __restrict__ cw, const float* __restrict__ cb,
    const float* __restrict__ w1, const float* __restrict__ b1,
    const float* __restrict__ w2, const float* __restrict__ b2,
    const float* __restrict__ sit, const float* __restrict__ gbal,
    float* __restrict__ out)
{
    __shared__ float ewt[16 * 132];      // [col 0..15][d 0..127], cols>=4 are zero
    __shared__ float elog[8 * 32 * 5];   // per wave: 32 rows x (4 experts + pad)
    __shared__ float smw[660];           // cw[12] cb[4] b2[4] w1[384] b1[128] w2[128]
    float* s_cw = smw;
    float* s_cb = smw + 12;
    float* s_b2 = smw + 16;
    float* s_w1 = smw + 20;
    float* s_b1 = smw + 20 + 384;
    float* s_w2 = smw + 20 + 384 + 128;

    const int tid = threadIdx.x;

    for (int i = tid; i < 16 * 132; i += 256) ewt[i] = 0.f;
    __syncthreads();
    for (int i = tid; i < D_B * K_B; i += 256) {
        int dd = i >> 2, cc = i & 3;
        ewt[cc * 132 + dd] = ew[i];      // transpose: [col][d]
    }
    if (tid < 12) s_cw[tid] = cw[tid];
    if (tid < 4) { s_cb[tid] = cb[tid]; s_b2[tid] = b2[tid]; }
    for (int i = tid; i < 384; i += 256) s_w1[i] = w1[i];
    for (int i = tid; i < 128; i += 256) { s_b1[i] = b1[i]; s_w2[i] = w2[i]; }
    __syncthreads();

    const int wave  = tid >> 5;
    const int lane  = tid & 31;
    const int col   = lane & 15;   // M index for A frags / N index for B,C frags
    const int khalf = lane >> 4;

    const long rbase = (long)blockIdx.x * 256 + (long)wave * 32;

    // A fragment pointers: lane i<16 holds row rbase+i, K = 2h..2h+1 of step.
    const float* a0p = edge + (rbase + col) * D_B;
    const float* a1p = edge + (rbase + 16 + col) * D_B;
    const float* bp  = ewt + col * 132;

    v8f c0 = {};
    v8f c1 = {};
#pragma unroll 4
    for (int kk = 0; kk < 32; ++kk) {
        const int kb = kk * 4 + khalf * 2;
        v2f a0 = *(const v2f*)(a0p + kb);   // global_load_b64
        v2f a1 = *(const v2f*)(a1p + kb);
        v2f b  = *(const v2f*)(bp + kb);    // ds_load_b64
        c0 = __builtin_amdgcn_wmma_f32_16x16x4_f32(false, a0, false, b,
                                                   (short)0, c0, false, false);
        c1 = __builtin_amdgcn_wmma_f32_16x16x4_f32(false, a1, false, b,
                                                   (short)0, c1, false, false);
    }

    // Scatter C fragments: VGPR p on lane-half h = (row p+8h, col). Only the
    // 4 real expert columns are kept.
    float* eb = elog + wave * 160;
    if (col < 4) {
#pragma unroll
        for (int p = 0; p < 8; ++p) {
            const int rowt = p + 8 * khalf;
            eb[rowt * 5 + col]        = c0[p];
            eb[(16 + rowt) * 5 + col] = c1[p];
        }
    }
    __syncthreads();

    // Per-row tail: one lane per row.
    const long row = rbase + lane;
    const float el[4] = { eb[lane * 5 + 0], eb[lane * 5 + 1],
                          eb[lane * 5 + 2], eb[lane * 5 + 3] };
    const float s0 = scores[row * 3 + 0];
    const float s1 = scores[row * 3 + 1];
    const float s2 = scores[row * 3 + 2];
    const int n = (int)(row >> 10);   // L = 1024

    const float alpha = 1.0f / (1.0f + expf(-gbal[0]));

    float gl[4], eo[4];
#pragma unroll
    for (int k = 0; k < 4; ++k) {
        float cl = s_cb[k] + s0 * s_cw[0 * 4 + k] + s1 * s_cw[1 * 4 + k]
                 + s2 * s_cw[2 * 4 + k] + el[k];
        gl[k] = alpha * sit[n * 4 + k] + (1.0f - alpha) * cl;

        float acc = s_b2[k];
        const float* wk = s_w1 + k * 96;   // (S=3, H=32) block
        const float* bk = s_b1 + k * 32;
        const float* vk = s_w2 + k * 32;
#pragma unroll 4
        for (int h = 0; h < 32; ++h) {
            float t = bk[h] + s0 * wk[h] + s1 * wk[32 + h] + s2 * wk[64 + h];
            acc += gelu_exact(t) * vk[h];
        }
        eo[k] = acc;
    }

    const float m = fmaxf(fmaxf(gl[0], gl[1]), fmaxf(gl[2], gl[3]));
    const float e0 = expf(gl[0] - m), e1 = expf(gl[1] - m);
    const float e2 = expf(gl[2] - m), e3 = expf(gl[3] - m);
    const float inv = 1.0f / (e0 + e1 + e2 + e3);
    out[row] = (e0 * eo[0] + e1 * eo[1] + e2 * eo[2] + e3 * eo[3]) * inv;
}

extern "C" void kernel_launch(void* const* d_in, const int* in_sizes, int n_in,
                              void* d_out, int out_size, void* d_ws, size_t ws_size,
                              hipStream_t stream) {
    const float* scores    = (const float*)d_in[0];
    const float* veh_repr  = (const float*)d_in[1];
    const float* cust_repr = (const float*)d_in[2];
    const float* edge_emb  = (const float*)d_in[3];
    const float* expert_w1 = (const float*)d_in[4];
    const float* expert_b1 = (const float*)d_in[5];
    const float* expert_w2 = (const float*)d_in[6];
    const float* expert_b2 = (const float*)d_in[7];
    const float* sit_w1    = (const float*)d_in[8];
    const float* sit_b1    = (const float*)d_in[9];
    const float* sit_w2    = (const float*)d_in[10];
    const float* sit_b2    = (const float*)d_in[11];
    const float* cand_w    = (const float*)d_in[12];
    const float* cand_b    = (const float*)d_in[13];
    const float* edge_w    = (const float*)d_in[14];
    const float* gate_bal  = (const float*)d_in[15];

    float* sit_logits = (float*)d_ws;            // N*K floats
    float* outp = (float*)d_out;                 // (N,1,L) = N*L floats

    sit_gate_kernel<<<N_B, 128, 0, stream>>>(veh_repr, cust_repr,
                                             sit_w1, sit_b1, sit_w2, sit_b2,
                                             sit_logits);

    const int rows = N_B * L_B;                  // 524288, divisible by 256
    fuse_kernel<<<rows / 256, 256, 0, stream>>>(scores, edge_emb, edge_w,
                                                cand_w, cand_b,
                                                expert_w1, expert_b1,
                                                expert_w2, expert_b2,
                                                sit_logits, gate_bal, outp);
}